// PosteriorHiddenTreeMarkovModel_18614388261460
// MI455X (gfx1250) — compile-verified
//
#include <hip/hip_runtime.h>

// ---------------- problem constants ----------------
#define BRN   3
#define DEPTH 7
#define NTREE 64
#define NPT   3280            // nodes per tree = (3^8-1)/2
#define NDIM  (NTREE * NPT)   // 209920
#define CD    8               // C_DIM
#define GD    8               // NGEN
#define MSYM  128
#define LEAF0 1093            // OFF[7]
#define NLEAF 2187            // 3^7 per tree

// workspace layout (float offsets)
// state arrays use [node][g][c] layout: idx = node*64 + g*8 + c  (coalesced in c)
#define A_OFF      0
#define LOGA_OFF   1536
#define B_OFF      3072
#define PI_OFF     11264
#define LOGPI_OFF  11456
#define SP_OFF     11648
#define LOGSP_OFF  11672
#define BETA_OFF   16384
#define STATE_SZ   (NDIM * 64)          // 13,434,880 floats
#define TBETA_OFF  (BETA_OFF + STATE_SZ)
#define EPS_OFF    (TBETA_OFF + STATE_SZ)
#define ELL_OFF    (EPS_OFF + STATE_SZ)

typedef float v2f __attribute__((ext_vector_type(2)));
typedef float v8f __attribute__((ext_vector_type(8)));

union V8U { v8f v; float f[8]; };

// ---------------- parameter softmax ----------------
__global__ __launch_bounds__(256)
void phtm_param_softmax(const float* __restrict__ lam_A, const float* __restrict__ lam_B,
                        const float* __restrict__ lam_Pi, const float* __restrict__ lam_SP,
                        float* __restrict__ ws) {
  const int tid = threadIdx.x;
  // A: softmax over cp (axis 0), 192 columns (cc,b,g)
  if (tid < 192) {
    const int g = tid & 7, b = (tid >> 3) % 3, cc = tid / 24;
    float w[CD], mx = -1e30f;
    for (int cp = 0; cp < CD; ++cp) {
      w[cp] = lam_A[((cp * 8 + cc) * 3 + b) * 8 + g];
      mx = fmaxf(mx, w[cp]);
    }
    float s = 0.f;
    for (int cp = 0; cp < CD; ++cp) { w[cp] -= mx; s += expf(w[cp]); }
    const float ls = logf(s), inv = 1.f / s;
    for (int cp = 0; cp < CD; ++cp) {
      const int idx = ((cp * 8 + cc) * 3 + b) * 8 + g;
      ws[A_OFF + idx]    = expf(w[cp]) * inv;
      ws[LOGA_OFF + idx] = w[cp] - ls;
    }
  }
  // B: softmax over m (axis 1), 64 columns (c,g), streaming passes
  if (tid < 64) {
    const int g = tid & 7, c = tid >> 3;
    float mx = -1e30f;
    for (int m = 0; m < MSYM; ++m) mx = fmaxf(mx, lam_B[(c * 128 + m) * 8 + g]);
    float s = 0.f;
    for (int m = 0; m < MSYM; ++m) s += expf(lam_B[(c * 128 + m) * 8 + g] - mx);
    const float inv = 1.f / s;
    for (int m = 0; m < MSYM; ++m)
      ws[B_OFF + (c * 128 + m) * 8 + g] = expf(lam_B[(c * 128 + m) * 8 + g] - mx) * inv;
  }
  // Pi: softmax over c (axis 0), 24 columns (b,g)
  if (tid < 24) {
    const int g = tid & 7, b = tid >> 3;
    float w[CD], mx = -1e30f;
    for (int c = 0; c < CD; ++c) { w[c] = lam_Pi[(c * 3 + b) * 8 + g]; mx = fmaxf(mx, w[c]); }
    float s = 0.f;
    for (int c = 0; c < CD; ++c) { w[c] -= mx; s += expf(w[c]); }
    const float ls = logf(s), inv = 1.f / s;
    for (int c = 0; c < CD; ++c) {
      const int idx = (c * 3 + b) * 8 + g;
      ws[PI_OFF + idx]    = expf(w[c]) * inv;
      ws[LOGPI_OFF + idx] = w[c] - ls;
    }
  }
  // SP: softmax over b (axis 0), 8 columns (g)
  if (tid < 8) {
    const int g = tid;
    float w[BRN], mx = -1e30f;
    for (int b = 0; b < BRN; ++b) { w[b] = lam_SP[b * 8 + g]; mx = fmaxf(mx, w[b]); }
    float s = 0.f;
    for (int b = 0; b < BRN; ++b) { w[b] -= mx; s += expf(w[b]); }
    const float ls = logf(s), inv = 1.f / s;
    for (int b = 0; b < BRN; ++b) {
      ws[SP_OFF + b * 8 + g]    = expf(w[b]) * inv;
      ws[LOGSP_OFF + b * 8 + g] = w[b] - ls;
    }
  }
}

// ---------------- leaf beta init ----------------
__global__ __launch_bounds__(256)
void phtm_leaf_init(float* __restrict__ ws, const int* __restrict__ x) {
  const int idx = blockIdx.x * 256 + threadIdx.x;   // NTREE*NLEAF*GD threads exactly
  const int g = idx & 7, li = idx >> 3;
  const int tree = li / NLEAF, j = li - tree * NLEAF;
  const int loc = LEAF0 + j, node = tree * NPT + loc, b = j % 3;
  const int xv = x[node];
  const float* Pi = ws + PI_OFF;
  const float* Bm = ws + B_OFF;
  float v[CD], s = 0.f;
  for (int c = 0; c < CD; ++c) {
    v[c] = Pi[(c * 3 + b) * 8 + g] * Bm[(c * 128 + xv) * 8 + g];
    s += v[c];
  }
  const float inv = 1.f / s;
  float* bout = ws + BETA_OFF + node * 64 + g * 8;   // [node][g][c], contiguous in c
  for (int c = 0; c < CD; ++c) bout[c] = v[c] * inv;
}

// ---------------- upward (beta/t_beta) level, WMMA ----------------
__global__ __launch_bounds__(128)
void phtm_upward_level(float* __restrict__ ws, const int* __restrict__ x,
                       int childBase, int parBase, int parsPerTree) {
  const int lane  = threadIdx.x & 31;
  const int tile  = blockIdx.x * 4 + (threadIdx.x >> 5);
  const int tileBase = tile * 16;
  const int lid   = lane & 15;
  const int khalf = lane >> 4;              // K-half of A/B operand
  const float colsel = (lid < 8) ? 1.0f : 0.0f;
  const int cp = lid & 7;

  const float* A_sm = ws + A_OFF;
  const float* Bm   = ws + B_OFF;
  const float* SPt  = ws + SP_OFF;
  float* beta  = ws + BETA_OFF;
  float* tbeta = ws + TBETA_OFF;

  // A-operand row parent for this lane (rows = 16 parents)
  const int q = tileBase + lid;
  const int tree = q / parsPerTree;
  const int ploc = parBase + (q - tree * parsPerTree);
  const int cnode0 = tree * NPT + childBase + 3 * (ploc - parBase);

  // D-layout row-group parents handled by this lane in the post phase
  const int prow0 = khalf * 8;
  int nodes_r[8], xv_r[8];
  for (int r = 0; r < 8; ++r) {
    const int pq = tileBase + prow0 + r;
    const int tr = pq / parsPerTree;
    const int pl = parBase + (pq - tr * parsPerTree);
    nodes_r[r] = tr * NPT + pl;
    xv_r[r] = x[nodes_r[r]];
  }

  for (int g = 0; g < GD; ++g) {
    v8f acc = {0.f, 0.f, 0.f, 0.f, 0.f, 0.f, 0.f, 0.f};
    for (int k = 0; k < BRN; ++k) {
      // child beta, [node][g][c] layout: contiguous pair loads (b64)
      const float* bptr = beta + (cnode0 + k) * 64 + g * 8;
      const v2f a1 = *(const v2f*)(bptr + khalf * 2);
      const v2f a2 = *(const v2f*)(bptr + 4 + khalf * 2);
      const float spv = SPt[k * 8 + g];
      // B operand: M1[cc][cp] = SP*A[cp][cc][k][g], column = cp(=lid)
      v2f b1, b2;
      b1.x = colsel * spv * A_sm[((cp * 8 + (khalf * 2 + 0)) * 3 + k) * 8 + g];
      b1.y = colsel * spv * A_sm[((cp * 8 + (khalf * 2 + 1)) * 3 + k) * 8 + g];
      b2.x = colsel * spv * A_sm[((cp * 8 + (4 + khalf * 2 + 0)) * 3 + k) * 8 + g];
      b2.y = colsel * spv * A_sm[((cp * 8 + (4 + khalf * 2 + 1)) * 3 + k) * 8 + g];
      acc = __builtin_amdgcn_wmma_f32_16x16x4_f32(false, a1, false, b1, (short)0, acc, false, false);
      acc = __builtin_amdgcn_wmma_f32_16x16x4_f32(false, a2, false, b2, (short)0, acc, false, false);
    }
    // post: lane holds t_beta[parent prow0+r, cp=lid] in acc[r]
    V8U u; u.v = acc;
    // coalesced t_beta stores: lanes 0..7 write 8 consecutive floats per r
    if (lid < 8)
      for (int r = 0; r < 8; ++r)
        tbeta[nodes_r[r] * 64 + g * 8 + lid] = u.f[r];
    float bu[8], srow[8];
    for (int r = 0; r < 8; ++r) {
      bu[r] = colsel * u.f[r] * Bm[(cp * 128 + xv_r[r]) * 8 + g];
      srow[r] = bu[r];
    }
    for (int m = 1; m < 8; m <<= 1)
      for (int r = 0; r < 8; ++r) srow[r] += __shfl_xor(srow[r], m, 32);
    if (lid < 8)
      for (int r = 0; r < 8; ++r)
        beta[nodes_r[r] * 64 + g * 8 + lid] = bu[r] / srow[r];
  }
}

// ---------------- root eps init, ell[root]=0 ----------------
__global__ __launch_bounds__(256)
void phtm_roots_init(float* __restrict__ ws) {
  const int idx = blockIdx.x * 256 + threadIdx.x;   // 4096 threads
  if (idx < NTREE * 64) {
    const int t = idx >> 6, gc = idx & 63;
    const int node = t * NPT;
    ws[EPS_OFF + node * 64 + gc] = ws[BETA_OFF + node * 64 + gc];
  }
  if (idx < NTREE * GD) {
    const int t = idx >> 3, g = idx & 7;
    ws[ELL_OFF + t * NPT * 8 + g] = 0.f;
  }
}

// ---------------- downward (eps/ell) level, WMMA ----------------
__global__ __launch_bounds__(128)
void phtm_downward_level(float* __restrict__ ws,
                         int childBase, int parBase, int parsPerTree) {
  const int lane  = threadIdx.x & 31;
  const int tile  = blockIdx.x * 4 + (threadIdx.x >> 5);
  const int tileBase = tile * 16;
  const int lid   = lane & 15;
  const int khalf = lane >> 4;
  const float colsel = (lid < 8) ? 1.0f : 0.0f;
  const int cc = lid & 7;

  const float* A_sm = ws + A_OFF;
  const float* lgA  = ws + LOGA_OFF;
  const float* SPt  = ws + SP_OFF;
  const float* lSP  = ws + LOGSP_OFF;
  const float* beta = ws + BETA_OFF;
  const float* tbet = ws + TBETA_OFF;
  float* eps = ws + EPS_OFF;
  float* ell = ws + ELL_OFF;

  // A-operand parent for this lane
  const int q = tileBase + lid;
  const int tree = q / parsPerTree;
  const int ploc = parBase + (q - tree * parsPerTree);
  const int pnode = tree * NPT + ploc;

  // D-layout row-group: child node base (pos k added in loop)
  const int prow0 = khalf * 8;
  int cnb_r[8];
  for (int r = 0; r < 8; ++r) {
    const int pq = tileBase + prow0 + r;
    const int tr = pq / parsPerTree;
    const int pl = parBase + (pq - tr * parsPerTree);
    cnb_r[r] = tr * NPT + childBase + 3 * (pl - parBase);
  }

  for (int g = 0; g < GD; ++g) {
    // w[cp] = eps[pa,cp,g] / t_beta[pa,cp,g] ; contiguous b64 pair loads
    const float* ep = eps  + pnode * 64 + g * 8;
    const float* tb = tbet + pnode * 64 + g * 8;
    const v2f epv1 = *(const v2f*)(ep + khalf * 2);
    const v2f tbv1 = *(const v2f*)(tb + khalf * 2);
    const v2f epv2 = *(const v2f*)(ep + 4 + khalf * 2);
    const v2f tbv2 = *(const v2f*)(tb + 4 + khalf * 2);
    const v2f aw1 = epv1 / tbv1;
    const v2f aw2 = epv2 / tbv2;

    for (int k = 0; k < BRN; ++k) {
      const float spv = SPt[k * 8 + g];
      const float lspv = lSP[k * 8 + g];
      const int c0 = khalf * 2 + 0, c1 = khalf * 2 + 1;
      const int c2 = 4 + khalf * 2 + 0, c3 = 4 + khalf * 2 + 1;
      const int i0 = ((c0 * 8 + cc) * 3 + k) * 8 + g;
      const int i1 = ((c1 * 8 + cc) * 3 + k) * 8 + g;
      const int i2 = ((c2 * 8 + cc) * 3 + k) * 8 + g;
      const int i3 = ((c3 * 8 + cc) * 3 + k) * 8 + g;
      const float a0 = A_sm[i0], a1v = A_sm[i1], a2v = A_sm[i2], a3v = A_sm[i3];
      v2f be1, be2, bl1, bl2;
      be1.x = colsel * spv * a0;          be1.y = colsel * spv * a1v;
      be2.x = colsel * spv * a2v;         be2.y = colsel * spv * a3v;
      bl1.x = be1.x * lgA[i0];            bl1.y = be1.y * lgA[i1];
      bl2.x = be2.x * lgA[i2];            bl2.y = be2.y * lgA[i3];

      v8f accE = {0.f,0.f,0.f,0.f,0.f,0.f,0.f,0.f};
      v8f accL = {0.f,0.f,0.f,0.f,0.f,0.f,0.f,0.f};
      accE = __builtin_amdgcn_wmma_f32_16x16x4_f32(false, aw1, false, be1, (short)0, accE, false, false);
      accE = __builtin_amdgcn_wmma_f32_16x16x4_f32(false, aw2, false, be2, (short)0, accE, false, false);
      accL = __builtin_amdgcn_wmma_f32_16x16x4_f32(false, aw1, false, bl1, (short)0, accL, false, false);
      accL = __builtin_amdgcn_wmma_f32_16x16x4_f32(false, aw2, false, bl2, (short)0, accL, false, false);

      V8U ue, ul; ue.v = accE; ul.v = accL;
      float ev[8], es[8], ls[8];
      for (int r = 0; r < 8; ++r) {
        const int cn = cnb_r[r] + k;
        const float bval = colsel * beta[cn * 64 + g * 8 + cc];   // coalesced read
        ev[r] = bval * ue.f[r];
        es[r] = ev[r];
        ls[r] = bval * ul.f[r];
      }
      // coalesced eps stores, single guarded block
      if (lid < 8)
        for (int r = 0; r < 8; ++r)
          eps[(cnb_r[r] + k) * 64 + g * 8 + lid] = ev[r];
      for (int m = 1; m < 8; m <<= 1)
        for (int r = 0; r < 8; ++r) {
          es[r] += __shfl_xor(es[r], m, 32);
          ls[r] += __shfl_xor(ls[r], m, 32);
        }
      if (lid == 0)
        for (int r = 0; r < 8; ++r)
          ell[(cnb_r[r] + k) * 8 + g] = ls[r] + es[r] * lspv;
    }
  }
}

// ---------------- final per-tree reduction ----------------
__global__ __launch_bounds__(256)
void phtm_final_reduce(const float* __restrict__ ws, const int* __restrict__ x,
                       float* __restrict__ out) {
  __shared__ float sdata[256];
  const int tree = blockIdx.x;
  const int tid = threadIdx.x;
  const int g = tid & 7, chunk = tid >> 3;
  const float* Bm   = ws + B_OFF;
  const float* lPi  = ws + LOGPI_OFF;
  const float* eps  = ws + EPS_OFF;
  const float* ell  = ws + ELL_OFF;
  float local = 0.f;
  for (int loc = chunk; loc < NPT; loc += 32) {
    const int node = tree * NPT + loc;
    float v = ell[node * 8 + g];
    const int xv = x[node];
    const float* ev = eps + node * 64 + g * 8;   // contiguous 8 floats
    float s = 0.f;
    for (int c = 0; c < CD; ++c)
      s += ev[c] * Bm[(c * 128 + xv) * 8 + g];
    v += s;
    if (loc >= LEAF0) {
      const int b = (loc - LEAF0) % 3;
      float s2 = 0.f;
      for (int c = 0; c < CD; ++c)
        s2 += ev[c] * lPi[(c * 3 + b) * 8 + g];
      v += s2;
    }
    local += v;
  }
  sdata[tid] = local;
  __syncthreads();
  for (int s = 128; s >= 8; s >>= 1) {
    if (tid < s) sdata[tid] += sdata[tid + s];
    __syncthreads();
  }
  if (tid < 8) out[tree * 8 + tid] = -sdata[tid];
}

// ---------------- launch ----------------
extern "C" void kernel_launch(void* const* d_in, const int* in_sizes, int n_in,
                              void* d_out, int out_size, void* d_ws, size_t ws_size,
                              hipStream_t stream) {
  (void)in_sizes; (void)n_in; (void)out_size; (void)ws_size;
  const float* lam_A  = (const float*)d_in[0];
  const float* lam_B  = (const float*)d_in[1];
  const float* lam_Pi = (const float*)d_in[2];
  const float* lam_SP = (const float*)d_in[3];
  const int*   x      = (const int*)d_in[4];
  float* ws  = (float*)d_ws;
  float* out = (float*)d_out;

  static const int OFFS[9] = {0, 1, 4, 13, 40, 121, 364, 1093, 3280};
  static const int P3[8]   = {1, 3, 9, 27, 81, 243, 729, 2187};

  phtm_param_softmax<<<1, 256, 0, stream>>>(lam_A, lam_B, lam_Pi, lam_SP, ws);
  phtm_leaf_init<<<(NTREE * NLEAF * GD) / 256, 256, 0, stream>>>(ws, x);

  for (int d = DEPTH; d >= 1; --d) {
    const int pars = P3[d - 1];
    const int tiles = (NTREE * pars) / 16;       // always a multiple of 4
    phtm_upward_level<<<tiles / 4, 128, 0, stream>>>(ws, x, OFFS[d], OFFS[d - 1], pars);
  }
  phtm_roots_init<<<16, 256, 0, stream>>>(ws);
  for (int d = 1; d <= DEPTH; ++d) {
    const int pars = P3[d - 1];
    const int tiles = (NTREE * pars) / 16;
    phtm_downward_level<<<tiles / 4, 128, 0, stream>>>(ws, OFFS[d], OFFS[d - 1], pars);
  }
  phtm_final_reduce<<<NTREE, 256, 0, stream>>>(ws, x, out);
}